// AtomicOrbitals_12893491823120
// MI455X (gfx1250) — compile-verified
//
#include <hip/hip_runtime.h>

// ---------------------------------------------------------------------------
// AtomicOrbitals forward on gfx1250.
//
// ao[row, o] = sum_s X[row, s] * M[s, o]
//   X[row, s] = r^n * exp(-alpha*r^2) * x^kx * y^ky * z^kz      (row = b*64+e)
//   M[s, o]   = (index_ctr[s] == o) ? norm_cst[s]*bas_coeffs[s] : 0
//
// The scatter-add contraction is expressed as a dense f32 GEMM using
// V_WMMA_F32_16X16X4_F32, so output is written exactly once, no atomics.
// ---------------------------------------------------------------------------

typedef __attribute__((ext_vector_type(2))) float v2f;
typedef __attribute__((ext_vector_type(8))) float v8f;

#define NBATCH 1024
#define NELEC 64
#define NSH 16          // shells per atom
#define NBAS 256
#define NORB 160
#define NROWS (NBATCH * NELEC)   // 65536
#define ROWS_PER_BLK 64          // 4 row-tiles of 16
#define NTHREADS 320             // 10 wave32s -> 10 column tiles of 16 orbitals
#define XSTRIDE 260              // 256 + 4 pad: conflict-free, 8B-aligned pairs

// Build the (NBAS x NORB) one-hot contraction matrix with coefficients folded in.
__global__ void build_M_kernel(const float* __restrict__ norm_cst,
                               const float* __restrict__ bas_coeffs,
                               const int* __restrict__ index_ctr,
                               float* __restrict__ M) {
  int i = blockIdx.x * blockDim.x + threadIdx.x;
  if (i >= NBAS * NORB) return;
  int s = i / NORB;
  int o = i - s * NORB;
  M[i] = (index_ctr[s] == o) ? norm_cst[s] * bas_coeffs[s] : 0.0f;
}

__global__ __launch_bounds__(NTHREADS) void ao_wmma_kernel(
    const float* __restrict__ pos,          // (1024, 192) == flat (row, 3)
    const float* __restrict__ atom_coords,  // (16, 3)
    const float* __restrict__ bas_exp,      // (256,)
    const int*   __restrict__ bas_kx,
    const int*   __restrict__ bas_ky,
    const int*   __restrict__ bas_kz,
    const float* __restrict__ M,            // (256, 160)
    float*       __restrict__ out) {        // (65536, 160)
  __shared__ float Xs[ROWS_PER_BLK * XSTRIDE];   // 66560 B
  __shared__ float posS[ROWS_PER_BLK * 3];

  const int tid = threadIdx.x;
  const int rowBase = blockIdx.x * ROWS_PER_BLK;

  // Stage this block's electron positions. Flat index of pos(b, e*3+d) is
  // (b*64+e)*3 + d == row*3 + d, so this is a contiguous copy.
  for (int i = tid; i < ROWS_PER_BLK * 3; i += NTHREADS)
    posS[i] = pos[rowBase * 3 + i];
  __syncthreads();

  // ---- Phase 1: X[row][s] = r^n exp(-a r^2) * x^kx y^ky z^kz into LDS ----
  for (int t = tid; t < ROWS_PER_BLK * NBAS; t += NTHREADS) {
    int row = t >> 8;        // / NBAS
    int s = t & (NBAS - 1);
    int atom = s >> 4;       // / NSH
    float x = posS[row * 3 + 0] - atom_coords[atom * 3 + 0];
    float y = posS[row * 3 + 1] - atom_coords[atom * 3 + 1];
    float z = posS[row * 3 + 2] - atom_coords[atom * 3 + 2];
    float r2 = x * x + y * y + z * z;
    float r = __builtin_sqrtf(r2);
    int kx = bas_kx[s], ky = bas_ky[s], kz = bas_kz[s];
    int n = kx + ky + kz;
    float rn = 1.0f;
    for (int i = 0; i < n; ++i) rn *= r;         // r^n, n in [0,6]; 0^0 == 1
    float val = rn * __expf(-bas_exp[s] * r2);
    float hx = 1.0f; for (int i = 0; i < kx; ++i) hx *= x;
    float hy = 1.0f; for (int i = 0; i < ky; ++i) hy *= y;
    float hz = 1.0f; for (int i = 0; i < kz; ++i) hz *= z;
    Xs[row * XSTRIDE + s] = val * hx * hy * hz;
  }
  __syncthreads();

  // ---- Phase 2: 64x160 = X(64x256) @ M(256x160) via f32 WMMA ----
  const int wave = tid >> 5;           // 0..9 -> column tile
  const int lane = tid & 31;
  const int l16 = lane & 15;
  const int koff = (lane >> 4) << 1;   // lanes 0-15: K=k,k+1; lanes 16-31: K=k+2,k+3
  const int col = wave * 16 + l16;

  v8f acc0 = {}, acc1 = {}, acc2 = {}, acc3 = {};
  for (int k = 0; k < NBAS; k += 4) {
    // B fragment (4x16 tile of M), L2-resident constant matrix.
    v2f b;
    b.x = M[(k + koff) * NORB + col];
    b.y = M[(k + koff + 1) * NORB + col];
    // A fragments for the four 16-row tiles (8B-aligned, bank-conflict-free).
    v2f a0 = *(const v2f*)&Xs[(0 * 16 + l16) * XSTRIDE + k + koff];
    v2f a1 = *(const v2f*)&Xs[(1 * 16 + l16) * XSTRIDE + k + koff];
    v2f a2 = *(const v2f*)&Xs[(2 * 16 + l16) * XSTRIDE + k + koff];
    v2f a3 = *(const v2f*)&Xs[(3 * 16 + l16) * XSTRIDE + k + koff];
    acc0 = __builtin_amdgcn_wmma_f32_16x16x4_f32(false, a0, false, b, (short)0, acc0, false, false);
    acc1 = __builtin_amdgcn_wmma_f32_16x16x4_f32(false, a1, false, b, (short)0, acc1, false, false);
    acc2 = __builtin_amdgcn_wmma_f32_16x16x4_f32(false, a2, false, b, (short)0, acc2, false, false);
    acc3 = __builtin_amdgcn_wmma_f32_16x16x4_f32(false, a3, false, b, (short)0, acc3, false, false);
  }

  // D layout: VGPR j -> row M=j (lanes 0-15) / M=j+8 (lanes 16-31), N=lane&15.
  const int rhalf = (lane >> 4) << 3;  // 0 or 8
#pragma unroll
  for (int j = 0; j < 8; ++j) {
    out[(size_t)(rowBase + 0 * 16 + j + rhalf) * NORB + col] = acc0[j];
    out[(size_t)(rowBase + 1 * 16 + j + rhalf) * NORB + col] = acc1[j];
    out[(size_t)(rowBase + 2 * 16 + j + rhalf) * NORB + col] = acc2[j];
    out[(size_t)(rowBase + 3 * 16 + j + rhalf) * NORB + col] = acc3[j];
  }
}

extern "C" void kernel_launch(void* const* d_in, const int* in_sizes, int n_in,
                              void* d_out, int out_size, void* d_ws, size_t ws_size,
                              hipStream_t stream) {
  const float* pos         = (const float*)d_in[0];
  const float* atom_coords = (const float*)d_in[1];
  const float* bas_exp     = (const float*)d_in[2];
  const float* bas_coeffs  = (const float*)d_in[3];
  const float* norm_cst    = (const float*)d_in[4];
  const int*   bas_kx      = (const int*)d_in[5];
  const int*   bas_ky      = (const int*)d_in[6];
  const int*   bas_kz      = (const int*)d_in[7];
  const int*   index_ctr   = (const int*)d_in[8];
  // d_in[9] = norb (160), fixed at compile time.

  float* M = (float*)d_ws;  // 256*160*4 = 160 KB scratch

  build_M_kernel<<<(NBAS * NORB + 255) / 256, 256, 0, stream>>>(
      norm_cst, bas_coeffs, index_ctr, M);

  ao_wmma_kernel<<<NROWS / ROWS_PER_BLK, NTHREADS, 0, stream>>>(
      pos, atom_coords, bas_exp, bas_kx, bas_ky, bas_kz, M, (float*)d_out);
}